// BKTNN_76596446757101
// MI455X (gfx1250) — compile-verified
//
#include <hip/hip_runtime.h>
#include <stdint.h>

// ---------------------------------------------------------------------------
// BKT forward scan, MI455X (gfx1250).
// Bandwidth-bound (~403 MB @ 23.3 TB/s => ~17us floor). No matmul structure,
// so the CDNA5 path used is async global->LDS DMA (ASYNCcnt) + wave32 layout.
// One thread = one student; y tiles staged coalesced through LDS, double
// buffered; recurrence kept in registers (fma + v_rcp_f32); outputs written
// with non-temporal b128 stores (write-once data, don't pollute L2).
// ---------------------------------------------------------------------------

#define BKT_EPS 1e-6f

constexpr int T_LEN   = 512;            // timesteps (reference T)
constexpr int TCHUNK  = 64;             // timesteps per LDS tile
constexpr int NCHUNK  = T_LEN / TCHUNK; // 8
constexpr int ROWS    = 256;            // students per block (8 waves)
constexpr int RSTRIDE = TCHUNK + 4;     // 68 dwords: 272B rows, 16B aligned,
                                        // 2-way LDS bank conflict only

typedef int   v4i  __attribute__((vector_size(16)));   // for async builtin
typedef float v4f  __attribute__((ext_vector_type(4)));
typedef int   v4si __attribute__((ext_vector_type(4)));

// ---- CDNA5 async global->LDS helpers --------------------------------------
__device__ __forceinline__ void async_load_b128(const int* gsrc, int* ldst) {
#if __has_builtin(__builtin_amdgcn_global_load_async_to_lds_b128)
  // signature (probe-confirmed): (v4i as(1)*, v4i as(3)*, Ii offset, Ii cpol)
  __builtin_amdgcn_global_load_async_to_lds_b128(
      (__attribute__((address_space(1))) v4i*)gsrc,
      (__attribute__((address_space(3))) v4i*)ldst,
      /*imm offset=*/0, /*cpol=*/0);
#else
  unsigned loff = (unsigned)(__SIZE_TYPE__)
      (__attribute__((address_space(3))) void*)ldst;
  asm volatile("global_load_async_to_lds_b128 %0, %1, off"
               :: "v"(loff), "v"(gsrc) : "memory");
#endif
}

__device__ __forceinline__ void wait_async0() {
#if __has_builtin(__builtin_amdgcn_s_wait_asynccnt)
  __builtin_amdgcn_s_wait_asynccnt(0);
#else
  asm volatile("s_wait_asynccnt 0" ::: "memory");
#endif
}

__device__ __forceinline__ float fast_sigmoid(float x) {
  return 1.0f / (1.0f + __expf(-x));
}

// ---------------------------------------------------------------------------
__global__ __launch_bounds__(ROWS)
void bkt_forward_kernel(const int*   __restrict__ X,
                        const int*   __restrict__ y,
                        const float* __restrict__ learn_w,
                        const float* __restrict__ guess_w,
                        const float* __restrict__ slip_w,
                        const float* __restrict__ prior_w,
                        float*       __restrict__ outC,
                        float*       __restrict__ outL) {
  __shared__ int ybuf[2][ROWS * RSTRIDE];  // 2 x 68KB double buffer

  const int tid = threadIdx.x;
  const int b   = blockIdx.x * ROWS + tid;
  const long long yblk = (long long)blockIdx.x * ROWS * T_LEN;

  // ---- per-student parameters: coalesced v4 gather of X, then 4 lookups
  const v4si xi = *(const v4si*)(X + 4 * (long long)b);
  const float l = fast_sigmoid(learn_w[xi.x]);
  const float g = fast_sigmoid(guess_w[xi.y]);
  const float s = fast_sigmoid(slip_w[xi.z]);
  float latent  = fast_sigmoid(prior_w[xi.w]);   // initial mastery prob
  const float oms = 1.0f - s;
  const float omg = 1.0f - g;

  // ---- coalesced async-DMA mapping: 16 lanes cover one 256B row of the tile
  const int urow = tid >> 4;          // 0..15
  const int ucol = (tid & 15) << 2;   // dword col: 0,4,...,60

  // prefetch chunk 0 into buffer 0
  {
    const int* gb = y + yblk + ucol;
#pragma unroll
    for (int k = 0; k < ROWS / 16; ++k) {
      const int row = urow + (k << 4);
      async_load_b128(gb + (long long)row * T_LEN,
                      &ybuf[0][row * RSTRIDE + ucol]);
    }
  }

  float* cBase = outC + (long long)b * T_LEN;
  float* lBase = outL + (long long)b * T_LEN;

  for (int c = 0; c < NCHUNK; ++c) {
    wait_async0();       // own wave's DMA for chunk c complete
    __syncthreads();     // all waves' DMA visible; prev compute done

    if (c + 1 < NCHUNK) {  // kick DMA for next chunk into the other buffer
      const long long coff = yblk + (long long)(c + 1) * TCHUNK + ucol;
      const int nb = (c + 1) & 1;
#pragma unroll
      for (int k = 0; k < ROWS / 16; ++k) {
        const int row = urow + (k << 4);
        async_load_b128(y + coff + (long long)row * T_LEN,
                        &ybuf[nb][row * RSTRIDE + ucol]);
      }
    }

    const int* myrow = &ybuf[c & 1][tid * RSTRIDE];
    const int tbase  = c * TCHUNK;

#pragma unroll 4
    for (int t4 = 0; t4 < TCHUNK; t4 += 4) {
      const v4si yv = *(const v4si*)(myrow + t4);  // ds_load_b128
      v4f cv, lv;

      // one BKT step: record (correct, latent) then advance latent
#define BKT_STEP(YV, COUT, LOUT)                                   \
      {                                                            \
        const float a    = latent * oms;            /* L(1-s)   */ \
        const float notl = 1.0f - latent;                          \
        const float cor  = fmaf(notl, g, a);                       \
        const float k1   = a * __builtin_amdgcn_rcpf(cor);         \
        const float bz   = latent * s;              /* L*s      */ \
        const float dz   = fmaf(notl, omg, bz);                    \
        const float k0   = bz * __builtin_amdgcn_rcpf(dz);         \
        const float k    = ((YV) > 0) ? k1 : k0;                   \
        const float nxt  = fmaf(1.0f - k, l, k);                   \
        (COUT) = cor;                                              \
        (LOUT) = latent;                                           \
        latent = fminf(fmaxf(nxt, BKT_EPS), 1.0f - BKT_EPS);       \
      }

      BKT_STEP(yv.x, cv.x, lv.x);
      BKT_STEP(yv.y, cv.y, lv.y);
      BKT_STEP(yv.z, cv.z, lv.z);
      BKT_STEP(yv.w, cv.w, lv.w);
#undef BKT_STEP

      // write-once streaming outputs: non-temporal b128 stores
      __builtin_nontemporal_store(cv, (v4f*)(cBase + tbase + t4));
      __builtin_nontemporal_store(lv, (v4f*)(lBase + tbase + t4));
    }
    // no trailing barrier needed: next iteration's wait+barrier orders
    // buffer reuse against all waves' reads of this tile.
  }
}

// ---------------------------------------------------------------------------
extern "C" void kernel_launch(void* const* d_in, const int* in_sizes, int n_in,
                              void* d_out, int out_size, void* d_ws, size_t ws_size,
                              hipStream_t stream) {
  (void)n_in; (void)out_size; (void)d_ws; (void)ws_size;

  const int*   X  = (const int*)d_in[0];     // (B,4) int32
  const int*   y  = (const int*)d_in[1];     // (B,T) int32
  const float* lw = (const float*)d_in[2];   // (N,1)
  const float* gw = (const float*)d_in[3];
  const float* sw = (const float*)d_in[4];
  const float* pw = (const float*)d_in[5];

  const int B = in_sizes[0] / 4;             // 65536
  float* outC = (float*)d_out;                         // corrects (B,T)
  float* outL = (float*)d_out + (long long)B * T_LEN;  // latents  (B,T)

  dim3 grid(B / ROWS), block(ROWS);
  hipLaunchKernelGGL(bkt_forward_kernel, grid, block, 0, stream,
                     X, y, lw, gw, sw, pw, outC, outL);
}